// DeepSetTopK_54254026883184
// MI455X (gfx1250) — compile-verified
//
#include <hip/hip_runtime.h>

// Top-8 of each 128-f32 row (262,144 rows), sorted descending.
// Streaming problem: 142 MB total traffic -> ~6.1 us roofline at 23.3 TB/s.
//
// Strategy (MI455X / gfx1250):
//  * wave32: each wave owns 32 rows (16 KB tile).
//  * Stage tile HBM -> LDS with GLOBAL_LOAD_ASYNC_TO_LDS_B128 (fully coalesced
//    512B bursts, tracked by ASYNCcnt, no VGPR landing zone).
//  * s_wait_asynccnt 0, then each LANE streams its own row out of LDS with
//    ds_load_b128.
//  * FOUR independent branch-free sorted top-8 insertion networks (one per
//    float4 component) -> 4-way ILP so v_max_f32/v_min_f32 pairs VOPD
//    dual-issue and the dependent chain is 4x shorter than a single network.
//  * Final bitonic top-k merge: for sorted-desc 8-lists A,B the elementwise
//    m_i = max(a_i, b_{7-i}) is a bitonic sequence holding top-8 of A∪B;
//    a 12-CE bitonic network restores descending order. 3 merges per row.
//  * Rows padded to 528 B in LDS (16B-aligned for b128) -> 2-way bank
//    conflict instead of 32-way.

#define TOPK_K          8
#define ROWS_PER_WAVE   32
#define WAVES_PER_BLOCK 4
#define BLOCK_THREADS   (WAVES_PER_BLOCK * 32)
#define ROW_ELEMS       128
#define ROW_BYTES       512
#define LDS_ROW_BYTES   528                              // +16B pad per row
#define WAVE_LDS_BYTES  (ROWS_PER_WAVE * LDS_ROW_BYTES)  // 16896 B per wave

// Branchless insert of xv into a descending-sorted 8-chain (15 ops).
#define TOPK_INSERT(L, xv)                                              \
  do {                                                                  \
    float xx = (xv), tt;                                                \
    tt = fmaxf(L##0, xx); xx = fminf(L##0, xx); L##0 = tt;              \
    tt = fmaxf(L##1, xx); xx = fminf(L##1, xx); L##1 = tt;              \
    tt = fmaxf(L##2, xx); xx = fminf(L##2, xx); L##2 = tt;              \
    tt = fmaxf(L##3, xx); xx = fminf(L##3, xx); L##3 = tt;              \
    tt = fmaxf(L##4, xx); xx = fminf(L##4, xx); L##4 = tt;              \
    tt = fmaxf(L##5, xx); xx = fminf(L##5, xx); L##5 = tt;              \
    tt = fmaxf(L##6, xx); xx = fminf(L##6, xx); L##6 = tt;              \
    L##7 = fmaxf(L##7, xx);                                             \
  } while (0)

// Compare-exchange keeping max in x (2 ops).
#define CE(x, y)                                                        \
  do { float t_ = fmaxf(x, y); y = fminf(x, y); x = t_; } while (0)

// Merge sorted-desc list B into sorted-desc list A (top-8 of union -> A).
#define TOPK_MERGE(A, B)                                                \
  do {                                                                  \
    A##0 = fmaxf(A##0, B##7); A##1 = fmaxf(A##1, B##6);                 \
    A##2 = fmaxf(A##2, B##5); A##3 = fmaxf(A##3, B##4);                 \
    A##4 = fmaxf(A##4, B##3); A##5 = fmaxf(A##5, B##2);                 \
    A##6 = fmaxf(A##6, B##1); A##7 = fmaxf(A##7, B##0);                 \
    /* bitonic rebuild, distances 4,2,1 */                              \
    CE(A##0, A##4); CE(A##1, A##5); CE(A##2, A##6); CE(A##3, A##7);     \
    CE(A##0, A##2); CE(A##1, A##3); CE(A##4, A##6); CE(A##5, A##7);     \
    CE(A##0, A##1); CE(A##2, A##3); CE(A##4, A##5); CE(A##6, A##7);     \
  } while (0)

#define DECL8(L, init)                                                  \
  float L##0 = (init), L##1 = L##0, L##2 = L##0, L##3 = L##0,           \
        L##4 = L##0, L##5 = L##0, L##6 = L##0, L##7 = L##0

__global__ void __launch_bounds__(BLOCK_THREADS)
topk8_async_kernel(const float* __restrict__ x, float* __restrict__ out,
                   int blocksPerBatch /*256*/, int outRowElems /*2048*/) {
  extern __shared__ float smem[];

  const int lane = threadIdx.x & 31;
  const int wave = threadIdx.x >> 5;
  const int tileRow = (blockIdx.x * WAVES_PER_BLOCK + wave) * ROWS_PER_WAVE;

  const float* gbase = x + (size_t)tileRow * ROW_ELEMS;   // wave-uniform SGPR base
  float* swave = smem + (size_t)wave * (WAVE_LDS_BYTES / 4);

  // Base LDS byte address of this wave's staging buffer (32-bit LDS offset).
  unsigned ldsBase =
      (unsigned)(unsigned long long)((__attribute__((address_space(3))) float*)swave);

  // ---- Stage 1: async copy 32 rows x 512B, coalesced b128 per lane ----
#pragma unroll
  for (int j = 0; j < ROWS_PER_WAVE; ++j) {
    unsigned e = (unsigned)(j * ROW_BYTES) + (unsigned)lane * 16u; // byte in tile
    unsigned r = e >> 9;         // row within tile
    unsigned c = e & 511u;       // byte within row
    unsigned ldsAddr = ldsBase + r * LDS_ROW_BYTES + c;
    asm volatile("global_load_async_to_lds_b128 %0, %1, %2"
                 :
                 : "v"(ldsAddr), "v"(e), "s"(gbase)
                 : "memory");
  }

#if __has_builtin(__builtin_amdgcn_s_wait_asynccnt)
  __builtin_amdgcn_s_wait_asynccnt(0);
  asm volatile("" ::: "memory");
#else
  asm volatile("s_wait_asynccnt 0x0" ::: "memory");
#endif

  // ---- Stage 2: each lane streams its own (padded) row from LDS into ----
  // ---- four independent top-8 networks (4-way ILP, VOPD-friendly).    ----
  const float4* rowp =
      (const float4*)((const char*)swave + (size_t)lane * LDS_ROW_BYTES);

  const float ninf = -__builtin_inff();
  DECL8(a, ninf);
  DECL8(b, ninf);
  DECL8(c, ninf);
  DECL8(d, ninf);

#pragma unroll 4
  for (int i = 0; i < ROW_ELEMS / 4; ++i) {
    float4 v = rowp[i];          // ds_load_b128
    TOPK_INSERT(a, v.x);
    TOPK_INSERT(b, v.y);
    TOPK_INSERT(c, v.z);
    TOPK_INSERT(d, v.w);
  }

  // ---- Stage 3: bitonic top-8 merges: a <- a∪b, c <- c∪d, a <- a∪c ----
  TOPK_MERGE(a, b);
  TOPK_MERGE(c, d);
  TOPK_MERGE(a, c);

  // ---- Stage 4: store 8 sorted values, 32B per lane, coalesced ----
  const int R   = tileRow + lane;
  const int bb  = R / blocksPerBatch;
  const int blk = R - bb * blocksPerBatch;
  float* o = out + (size_t)bb * outRowElems + (size_t)blk * TOPK_K;
  ((float4*)o)[0] = make_float4(a0, a1, a2, a3);
  ((float4*)o)[1] = make_float4(a4, a5, a6, a7);
}

extern "C" void kernel_launch(void* const* d_in, const int* in_sizes, int n_in,
                              void* d_out, int out_size, void* d_ws, size_t ws_size,
                              hipStream_t stream) {
  (void)n_in; (void)d_ws; (void)ws_size; (void)out_size;
  const float* x = (const float*)d_in[0];
  float* out = (float*)d_out;

  const int n = in_sizes[0];                 // 1024*256*128
  const int rows = n / ROW_ELEMS;            // 262144
  const int rowsPerBlock = WAVES_PER_BLOCK * ROWS_PER_WAVE;  // 128
  const int grid = rows / rowsPerBlock;      // 2048
  const size_t shmem = (size_t)WAVES_PER_BLOCK * WAVE_LDS_BYTES;  // 67584 B

  topk8_async_kernel<<<grid, BLOCK_THREADS, shmem, stream>>>(
      x, out, /*blocksPerBatch=*/256, /*outRowElems=*/256 * TOPK_K);
}